// CascadeLayer_15556371546769
// MI455X (gfx1250) — compile-verified
//
#include <hip/hip_runtime.h>
#include <hip/hip_bf16.h>

#define DIM 128
#define ALPHA 4

typedef __attribute__((ext_vector_type(2))) float v2f;
typedef __attribute__((ext_vector_type(8))) float v8f;

// ---------------------------------------------------------------------------
// WMMA f32 GEMM: out[m,n] = act( sum_k A[m,k] * W[n,k] + bias[n] )
// A: [nrows, 128] row-major, W: [128, 128] row-major (out = A @ W^T).
// One 256-thread block = 8 waves; block b covers rows [16b, 16b+16),
// wave w covers output cols [16w, 16w+16). K swept in steps of 4 with
// V_WMMA_F32_16X16X4_F32 (exact f32 FMA path).
//
// Fragment layouts (CDNA5 ISA 7.12.2, wave32):
//   A 16x4 f32 : lane L<16 -> (M=L, K=0..1), lane>=16 -> (M=L-16, K=2..3)
//   B 4x16 f32 : lane L<16 -> (K=0..1, N=L), lane>=16 -> (K=2..3, N=L-16)
//   C/D 16x16  : vgpr r, lane L<16 -> (M=r, N=L); lane>=16 -> (M=r+8, N=L-16)
// ---------------------------------------------------------------------------
__global__ __launch_bounds__(256) void gemm_wmma_f32(
    const float* __restrict__ A, const float* __restrict__ W,
    const float* __restrict__ bias, float* __restrict__ out,
    int nrows, int do_relu)
{
    const int lane = threadIdx.x & 31;
    const int wave = threadIdx.x >> 5;     // 0..7 -> column tile
    const int m0   = blockIdx.x << 4;      // row-tile base
    if (m0 >= nrows) return;               // uniform per block
    const int n0   = wave << 4;
    const int half = lane >> 4;            // 0: K=0..1 / 1: K=2..3
    const int l    = lane & 15;

    const float* __restrict__ arow = A + (size_t)(m0 + l) * DIM + 2 * half;
    const float* __restrict__ brow = W + (size_t)(n0 + l) * DIM + 2 * half;

    v8f c = {};
#pragma unroll 8
    for (int k = 0; k < DIM; k += 4) {
        v2f a = *(const v2f*)(arow + k);
        v2f b = *(const v2f*)(brow + k);
        c = __builtin_amdgcn_wmma_f32_16x16x4_f32(
            /*neg_a=*/false, a, /*neg_b=*/false, b,
            /*c_mod=*/(short)0, c, /*reuse_a=*/false, /*reuse_b=*/false);
    }

    const float bv = bias ? bias[n0 + l] : 0.0f;
#pragma unroll
    for (int r = 0; r < 8; ++r) {
        const int m = m0 + r + 8 * half;
        float v = c[r] + bv;
        if (do_relu) v = fmaxf(v, 0.0f);
        out[(size_t)m * DIM + (n0 + l)] = v;
    }
}

// deg[col[e]] += w[e]
__global__ __launch_bounds__(256) void deg_kernel(
    const int* __restrict__ col, const float* __restrict__ w,
    float* __restrict__ deg, int nedges)
{
    const int e = blockIdx.x * blockDim.x + threadIdx.x;
    if (e < nedges) atomicAdd(deg + col[e], w[e]);
}

// dinv[n] = deg[n] > 0 ? rsqrt(deg[n]) : 0
__global__ __launch_bounds__(256) void dinv_kernel(
    const float* __restrict__ deg, float* __restrict__ dinv, int n)
{
    const int i = blockIdx.x * blockDim.x + threadIdx.x;
    if (i < n) {
        const float d = deg[i];
        dinv[i] = (d > 0.0f) ? __frsqrt_rn(d) : 0.0f;
    }
}

// One wave32 per edge: each lane handles 4 contiguous floats of the 128-dim
// row. acc[col[e], :] += (dinv[row]*w*dinv[col]) * hw[row[e], :]
__global__ __launch_bounds__(256) void scatter_kernel(
    const int* __restrict__ row, const int* __restrict__ col,
    const float* __restrict__ w, const float* __restrict__ dinv,
    const float* __restrict__ hw, float* __restrict__ acc, int nedges)
{
    const int e    = (int)((blockIdx.x * (size_t)blockDim.x + threadIdx.x) >> 5);
    const int lane = threadIdx.x & 31;
    if (e >= nedges) return;
    const int r = row[e];
    const int c = col[e];
    const float norm = dinv[r] * w[e] * dinv[c];
    if (norm == 0.0f) return;  // zero-degree endpoints contribute nothing
    const float4 v = *(const float4*)(hw + (size_t)r * DIM + lane * 4);
    float* dst = acc + (size_t)c * DIM + lane * 4;
    atomicAdd(dst + 0, norm * v.x);
    atomicAdd(dst + 1, norm * v.y);
    atomicAdd(dst + 2, norm * v.z);
    atomicAdd(dst + 3, norm * v.w);
}

// In place: out[i] = relu(out[i] + bias[i % 128])
__global__ __launch_bounds__(256) void bias_relu_kernel(
    float* __restrict__ out, const float* __restrict__ bias, int total)
{
    const int i = blockIdx.x * blockDim.x + threadIdx.x;
    if (i < total) out[i] = fmaxf(out[i] + bias[i & (DIM - 1)], 0.0f);
}

extern "C" void kernel_launch(void* const* d_in, const int* in_sizes, int n_in,
                              void* d_out, int out_size, void* d_ws, size_t ws_size,
                              hipStream_t stream) {
    const float* x      = (const float*)d_in[0];
    const int*   eidx   = (const int*)d_in[1];   // [ALPHA, 2, E]
    const float* eattr  = (const float*)d_in[2]; // [ALPHA, E]
    const float* lin_w  = (const float*)d_in[3];
    const float* lin_b  = (const float*)d_in[4];
    const float* conv_w = (const float*)d_in[5]; // [ALPHA, 128, 128]
    const float* conv_b = (const float*)d_in[6]; // [ALPHA, 128]
    float*       out    = (float*)d_out;         // [ALPHA+1, N, 128]

    const int N = in_sizes[0] / DIM;             // 100000
    const int E = in_sizes[2] / ALPHA;           // 1600000

    // Workspace carve-up: hw [N*128], deg [N], dinv [N]
    float* hw   = (float*)d_ws;
    float* deg  = hw + (size_t)N * DIM;
    float* dinv = deg + N;

    const int row_tiles = N / 16;                // 6250 (exact)
    const dim3 blk(256);

    // h0 = relu(x @ lin_w^T + lin_b)  -> slice 0
    gemm_wmma_f32<<<row_tiles, blk, 0, stream>>>(x, lin_w, lin_b, out, N, 1);

    for (int i = 0; i < ALPHA; ++i) {
        const int*   rowi = eidx + (size_t)i * 2 * E;
        const int*   coli = rowi + E;
        const float* wi   = eattr + (size_t)i * E;
        const float* h    = out + (size_t)i * N * DIM;
        float*       hn   = out + (size_t)(i + 1) * N * DIM;

        // deg / dinv
        hipMemsetAsync(deg, 0, (size_t)N * sizeof(float), stream);
        deg_kernel<<<(E + 255) / 256, blk, 0, stream>>>(coli, wi, deg, E);
        dinv_kernel<<<(N + 255) / 256, blk, 0, stream>>>(deg, dinv, N);

        // hw = h @ conv_w[i]^T (no bias/relu)
        gemm_wmma_f32<<<row_tiles, blk, 0, stream>>>(
            h, conv_w + (size_t)i * DIM * DIM, nullptr, hw, N, 0);

        // scatter-add into next slice, then bias + relu in place
        hipMemsetAsync(hn, 0, (size_t)N * DIM * sizeof(float), stream);
        {
            const size_t threads = (size_t)E * 32;
            const int grid = (int)((threads + 255) / 256);
            scatter_kernel<<<grid, blk, 0, stream>>>(rowi, coli, wi, dinv, hw, hn, E);
        }
        bias_relu_kernel<<<((size_t)N * DIM + 255) / 256, blk, 0, stream>>>(
            hn, conv_b + (size_t)i * DIM, N * DIM);
    }
}